// ExtrudeNet_52269751993077
// MI455X (gfx1250) — compile-verified
//
#include <hip/hip_runtime.h>
#include <hip/hip_bf16.h>
#include <math.h>

typedef float v2f __attribute__((ext_vector_type(2)));
typedef float v8f __attribute__((ext_vector_type(8)));

#define B_ 4
#define M_ 2048
#define K_ 64
#define C_ 64
#define P_ 28
#define NCURVE 64   // NSEG(4) * SAMPLE_RATE(16)

__device__ __forceinline__ v8f vmin8(v8f a, v8f b){
  v8f r;
#pragma unroll
  for (int i = 0; i < 8; ++i) r[i] = fminf(a[i], b[i]);
  return r;
}

// One block = one (b,k) primitive x 64-point tile of m. 128 threads = 4 wave32.
// Each wave handles 16 points. Per 16-edge/curve tile, THREE K=4 WMMAs:
//   d^2 : A row = [-2cx, -2cy, 1, cx^2+cy^2] , B col = [px, py, |p|^2, 1]
//   cr  : A row = [ay-by, bx-ax, ax*by-ay*bx, 0], B col = [px, py, 1, 0]
//   yt  : A row = [0, 1, -ay, 0]                , B col = [px, py, 1, 0]
// D column n = point n lives in lanes n / n+16 (8 rows each), so the min and
// the crossing-count winding run in-register per lane, no LDS in the loop.
__global__ __launch_bounds__(128) void sdf_kernel(
    const float* __restrict__ pts, const float* __restrict__ prim,
    float* __restrict__ sdf_out, float* __restrict__ dmin_out)
{
  __shared__ float4 s_c4[NCURVE];   // {cx, cy, cx^2+cy^2, 1}
  __shared__ float4 s_e4[NCURVE];   // {ay-by, bx-ax, ax*by-ay*bx, ay}
  const int b = blockIdx.x >> 6;
  const int k = blockIdx.x & 63;
  const float* Pp = prim + (size_t)b * P_ * K_ + k;   // param p at Pp[p*K_]
  const int tid = threadIdx.x;
  const float PI = 3.14159265358979323846f;

  // ---- rational Bezier curve (64 pts) into LDS ----
  if (tid < NCURVE) {
    int seg = tid >> 4;
    float t = (float)(tid & 15) * (1.0f / 16.0f);
    float theta = PI * 0.125f + atanf(tanf(PI * 0.125f) * (1.0f / 3.0f));
    float r0 = (PI * 0.5f) * (float)seg;
    float r1 = r0 + theta;
    float r2 = (PI * 0.5f) * (float)(seg + 1) - theta;
    float r3 = (PI * 0.5f) * (float)(seg + 1);
    float rad0 = fabsf(Pp[(8 + seg * 3 + 0) * K_]);
    float rad1 = fabsf(Pp[(8 + seg * 3 + 1) * K_]);
    float rad2 = fabsf(Pp[(8 + seg * 3 + 2) * K_]);
    int seg1 = (seg + 1) & 3;
    float rad3 = fabsf(Pp[(8 + seg1 * 3 + 0) * K_]);
    float P0x = cosf(r0) * rad0, P0y = sinf(r0) * rad0;
    float P1x = cosf(r1) * rad1, P1y = sinf(r1) * rad1;
    float P2x = cosf(r2) * rad2, P2y = sinf(r2) * rad2;
    float P3x = cosf(r3) * rad3, P3y = sinf(r3) * rad3;
    float w1 = fabsf(Pp[(20 + seg * 2 + 0) * K_]);
    float w2 = fabsf(Pp[(20 + seg * 2 + 1) * K_]);
    float u = 1.0f - t;
    float b0 = u * u * u, b1 = 3.0f * u * u * t, b2 = 3.0f * u * t * t, b3 = t * t * t;
    float inv = 1.0f / (b0 + w1 * b1 + w2 * b2 + b3);
    float cx = (P0x * b0 + P1x * (w1 * b1) + P2x * (w2 * b2) + P3x * b3) * inv;
    float cy = (P0y * b0 + P1y * (w1 * b1) + P2y * (w2 * b2) + P3y * b3) * inv;
    s_c4[tid] = make_float4(cx, cy, cx * cx + cy * cy, 1.0f);
  }

  // ---- pose (uniform per block -> scalar loads) ----
  float q0 = Pp[0], q1 = Pp[K_], q2 = Pp[2 * K_], q3 = Pp[3 * K_];
  float invn = 1.0f / sqrtf(q0 * q0 + q1 * q1 + q2 * q2 + q3 * q3 + 1e-8f);
  float qw = q0 * invn, vx = -q1 * invn, vy = -q2 * invn, vz = -q3 * invn;
  float tx = Pp[4 * K_], ty = Pp[5 * K_], tz = Pp[6 * K_];
  float habs = fabsf(Pp[7 * K_]);
  __syncthreads();

  // ---- edge descriptors for winding WMMAs ----
  if (tid < NCURVE) {
    float4 ca = s_c4[tid];
    float4 cb = s_c4[(tid + 1) & 63];
    s_e4[tid] = make_float4(ca.y - cb.y, cb.x - ca.x,
                            ca.x * cb.y - ca.y * cb.x, ca.y);
  }

  const int lane = tid & 31;
  const int wv = tid >> 5;
  const int row = lane & 15;
  const bool hi = lane >= 16;
  const int m = blockIdx.y * 64 + wv * 16 + row;

  // ---- rotate point into primitive frame ----
  const float* p3 = pts + ((size_t)b * M_ + m) * 3;
  float px0 = p3[0] - tx, py0 = p3[1] - ty, pz0 = p3[2] - tz;
  float t2x = 2.0f * (vy * pz0 - vz * py0);
  float t2y = 2.0f * (vz * px0 - vx * pz0);
  float t2z = 2.0f * (vx * py0 - vy * px0);
  float plx = px0 + qw * t2x + (vy * t2z - vz * t2y);
  float ply = py0 + qw * t2y + (vz * t2x - vx * t2z);
  float plz = pz0 + qw * t2z + (vx * t2y - vy * t2x);
  float px = plx, py = ply;
  float ss = px * px + py * py;
  __syncthreads();

  // point fragments (constant over tiles)
  // A layout: lanes0-15 v0=K0,v1=K1 ; lanes16-31 v0=K2,v1=K3 (rows = lane%16).
  // B layout: v0 = rows K0 / K2 ; v1 = rows K1 / K3 (cols = lane%16).
  v2f Bd; Bd.x = hi ? ss : px;   Bd.y = hi ? 1.0f : py;   // [px,py,ss,1]
  v2f Bw; Bw.x = hi ? 1.0f : px; Bw.y = hi ? 0.0f : py;   // [px,py,1,0]
  v8f z = {0.f, 0.f, 0.f, 0.f, 0.f, 0.f, 0.f, 0.f};
  v8f e = {3e38f, 3e38f, 3e38f, 3e38f, 3e38f, 3e38f, 3e38f, 3e38f};

  int wn = 0;
  int le0b = 0;       // bit t: aLE of first row (v=0) of tile t in this lane
  float cr7[4];
  int   le7[4];
#pragma unroll
  for (int tTile = 0; tTile < 4; ++tTile) {
    float4 cc = s_c4[tTile * 16 + row];
    float4 ee = s_e4[tTile * 16 + row];
    v2f Ad; Ad.x = hi ? 1.0f   : -2.0f * cc.x;  Ad.y = hi ? cc.z : -2.0f * cc.y;
    v2f Ac; Ac.x = hi ? ee.z   : ee.x;          Ac.y = hi ? 0.0f : ee.y;
    v2f Ay; Ay.x = hi ? -ee.w  : 0.0f;          Ay.y = hi ? 0.0f : 1.0f;
    v8f d  = __builtin_amdgcn_wmma_f32_16x16x4_f32(
        false, Ad, false, Bd, (short)0, z, false, false);
    v8f cr = __builtin_amdgcn_wmma_f32_16x16x4_f32(
        false, Ac, false, Bw, (short)0, z, false, false);
    v8f yd = __builtin_amdgcn_wmma_f32_16x16x4_f32(
        false, Ay, false, Bw, (short)0, z, false, false);
    e = vmin8(e, d);
    // my rows = edges tTile*16 + 8*hi + v, v = 0..7
    int le[8];
#pragma unroll
    for (int v = 0; v < 8; ++v) le[v] = (yd[v] >= 0.0f);
#pragma unroll
    for (int v = 0; v < 7; ++v) {
      int aLE = le[v], bLE = le[v + 1];
      float c = cr[v];
      wn += (aLE & (bLE ^ 1) & (c > 0.0f)) - ((aLE ^ 1) & bLE & (c < 0.0f));
    }
    le0b |= le[0] << tTile;
    cr7[tTile] = cr[7];
    le7[tTile] = le[7];
  }
  // boundary edges (v=7): next edge's aLE lives in the other half's le[0]
  int recvb = __shfl_xor(le0b, 16, 32);
#pragma unroll
  for (int tTile = 0; tTile < 4; ++tTile) {
    int nt = hi ? ((tTile + 1) & 3) : tTile;
    int bLE = (recvb >> nt) & 1;
    int aLE = le7[tTile];
    float c = cr7[tTile];
    wn += (aLE & (bLE ^ 1) & (c > 0.0f)) - ((aLE ^ 1) & bLE & (c < 0.0f));
  }
  wn += __shfl_xor(wn, 16, 32);

  // d2 min: 8 rows in-lane, 8 in lane^16
  float m01 = fminf(e[0], e[1]), m23 = fminf(e[2], e[3]);
  float m45 = fminf(e[4], e[5]), m67 = fminf(e[6], e[7]);
  float m8  = fminf(fminf(m01, m23), fminf(m45, m67));
  float d2min = fminf(m8, __shfl_xor(m8, 16, 32));

  float dminv = sqrtf(d2min + 1e-12f);
  float sdf2d = (wn != 0) ? -dminv : dminv;
  float dz = fabsf(plz) - habs;
  float po = fmaxf(sdf2d, 0.0f), dzo = fmaxf(dz, 0.0f);
  float sdf = fminf(fmaxf(sdf2d, dz), 0.0f) + sqrtf(po * po + dzo * dzo + 1e-12f);

  // full-exec store: lanes 0-15 write sdf, lanes 16-31 write dmin
  size_t o = ((size_t)b * M_ + m) * K_ + k;
  float* basep = hi ? dmin_out : sdf_out;
  float  outv  = hi ? dminv    : sdf;
  basep[o] = outv;
}

// One 64-thread block per (b,m). Thread c owns channel c.
// occ recomputed from sdf output (no workspace). Branchless two-pass softmax
// over K; union softmax reduced with wave shuffles.
__global__ __launch_bounds__(64) void combine_kernel(
    const float* __restrict__ sdf_in, const float* __restrict__ iw,
    const float* __restrict__ uw, const int* __restrict__ is_training,
    float* __restrict__ inter_out, float* __restrict__ occ_out)
{
  const int bm = blockIdx.x;
  const int b  = bm >> 11;            // M_ = 2048
  const int c  = threadIdx.x;
  __shared__ float occ_sh[K_];
  __shared__ float preu[C_];

  float sv = sdf_in[(size_t)bm * K_ + c];
  occ_sh[c] = 1.0f / (1.0f + __expf(150.0f * sv));
  const float* wp = iw + ((size_t)b * K_) * C_ + c;   // stride C_ over k, coalesced over c
  __builtin_prefetch(wp, 0, 3);
  int train = is_training[0];
  __syncthreads();

  float inter;
  if (train) {
    // softmax(-40*p) weighted sum over k, two-pass (branchless)
    float pmin = 3.0e38f;
#pragma unroll 4
    for (int kk = 0; kk < K_; ++kk) {
      float p = 1.0f - wp[(size_t)kk * C_] * (1.0f - occ_sh[kk]);
      pmin = fminf(pmin, p);
    }
    float s = 0.0f, ts = 0.0f;
#pragma unroll 4
    for (int kk = 0; kk < K_; ++kk) {
      float p = 1.0f - wp[(size_t)kk * C_] * (1.0f - occ_sh[kk]);
      float ee = __expf(-40.0f * (p - pmin));
      s += ee; ts += ee * p;
    }
    inter = ts / s;
  } else {
    inter = 3.0e38f;
#pragma unroll 4
    for (int kk = 0; kk < K_; ++kk) {
      float p = 1.0f - wp[(size_t)kk * C_] * (1.0f - occ_sh[kk]);
      inter = fminf(inter, p);
    }
  }
  inter_out[(size_t)bm * C_ + c] = inter;
  preu[c] = uw[b * C_ + c] * inter;
  __syncthreads();

  // union: softmax(40*pre_u) weighted sum over c, reduced by wave 0
  if (c < 32) {
    float a0 = preu[c], a1 = preu[c + 32];
    if (train) {
      float mx = fmaxf(a0, a1);
#pragma unroll
      for (int mask = 1; mask <= 16; mask <<= 1)
        mx = fmaxf(mx, __shfl_xor(mx, mask, 32));
      float e0 = __expf(40.0f * (a0 - mx));
      float e1 = __expf(40.0f * (a1 - mx));
      float s  = e0 + e1;
      float ts = e0 * a0 + e1 * a1;
#pragma unroll
      for (int mask = 1; mask <= 16; mask <<= 1) {
        s  += __shfl_xor(s,  mask, 32);
        ts += __shfl_xor(ts, mask, 32);
      }
      if (c == 0) occ_out[bm] = ts / s;
    } else {
      float mx = fmaxf(a0, a1);
#pragma unroll
      for (int mask = 1; mask <= 16; mask <<= 1)
        mx = fmaxf(mx, __shfl_xor(mx, mask, 32));
      if (c == 0) occ_out[bm] = mx;
    }
  }
}

extern "C" void kernel_launch(void* const* d_in, const int* in_sizes, int n_in,
                              void* d_out, int out_size, void* d_ws, size_t ws_size,
                              hipStream_t stream) {
  const float* pts   = (const float*)d_in[0];  // (B,M,3)
  const float* prim  = (const float*)d_in[1];  // (B,P,K)
  const float* iw    = (const float*)d_in[2];  // (B,K,C)
  const float* uw    = (const float*)d_in[3];  // (B,C)
  const int*   train = (const int*)d_in[4];    // scalar

  // outputs concatenated: occupancies (B*M), primitive_sdf (B*M*K),
  //                       inter (B*M*C), support_distances (B*M*K)
  float* occ_out   = (float*)d_out;
  float* sdf_out   = occ_out + (size_t)B_ * M_;
  float* inter_out = sdf_out + (size_t)B_ * M_ * K_;
  float* dmin_out  = inter_out + (size_t)B_ * M_ * C_;

  dim3 g1(B_ * K_, M_ / 64);
  sdf_kernel<<<g1, 128, 0, stream>>>(pts, prim, sdf_out, dmin_out);
  combine_kernel<<<B_ * M_, 64, 0, stream>>>(sdf_out, iw, uw, train,
                                             inter_out, occ_out);
  (void)in_sizes; (void)n_in; (void)out_size; (void)d_ws; (void)ws_size;
}